// SmallGCN_21655225106952
// MI455X (gfx1250) — compile-verified
//
#include <hip/hip_runtime.h>
#include <hip/hip_bf16.h>

// ---------------------------------------------------------------------------
// SmallGCN on gfx1250 (MI455X)
//   layer1: h = x @ W1            (fp32 WMMA 16x16x4)
//           agg[col] += h[row] * dinv[row]*dinv[col]   (wave-per-edge, f32 atomics, L2-resident)
//           hrelu = relu(agg + b1)
//   layer2: z = hrelu @ W2        (fp32 WMMA, 10 valid cols of a 16-tile)
//           out = b2 ; out[col] += z[row] * norm       (thread-per-edge)
// ---------------------------------------------------------------------------

typedef float v2f __attribute__((ext_vector_type(2)));
typedef float v8f __attribute__((ext_vector_type(8)));

#define N_NODES   100000
#define N_EDGES   1600000
#define IN_DIM    256
#define HIDDEN    128
#define N_CLASSES 10

// ---------------- degree / norm ----------------

__global__ void gcn_deg_kernel(const int* __restrict__ col, float* __restrict__ deg) {
    int e = blockIdx.x * blockDim.x + threadIdx.x;
    if (e < N_EDGES) {
        atomicAdd(&deg[col[e]], 1.0f);
    }
}

__global__ void gcn_dinv_kernel(float* __restrict__ deg_dinv) {
    int i = blockIdx.x * blockDim.x + threadIdx.x;
    if (i < N_NODES) {
        float d = deg_dinv[i];
        deg_dinv[i] = (d > 0.0f) ? rsqrtf(fmaxf(d, 1.0f)) : 0.0f;
    }
}

// ---------------- GEMM1: h[N,128] = x[N,256] @ W1[256,128] (fp32 WMMA) -------
// One wave per 16-row tile; 8 column tiles (8 x v8f accumulators).
// A 16x4 f32 fragment: lanes 0-15 -> M=lane, K={0,1}; lanes 16-31 -> M=lane-16, K={2,3}.
// B 4x16 f32 fragment: symmetric (N striped over lanes, K over VGPRs).

__global__ void gcn_gemm1_wmma(const float* __restrict__ x,
                               const float* __restrict__ W1,
                               float* __restrict__ h) {
    const int lane = threadIdx.x & 31;
    const int wave = threadIdx.x >> 5;
    const int tile = blockIdx.x * 8 + wave;       // 16-row tile
    const int nTiles = N_NODES / 16;              // 6250
    if (tile >= nTiles) return;                   // wave-uniform exit

    const int m    = lane & 15;
    const int koff = (lane >> 4) * 2;             // 0 or 2
    const int n    = lane & 15;

    v8f acc[8] = {};

    const float* xrow = x + (size_t)(tile * 16 + m) * IN_DIM;

    for (int k0 = 0; k0 < IN_DIM; k0 += 4) {
        v2f a = *(const v2f*)(xrow + k0 + koff);  // 8B load, reused across 8 B-tiles
        const float* Wk = W1 + (size_t)(k0 + koff) * HIDDEN + n;
#pragma unroll
        for (int t = 0; t < 8; ++t) {
            v2f b;
            b.x = Wk[t * 16];
            b.y = Wk[t * 16 + HIDDEN];
            acc[t] = __builtin_amdgcn_wmma_f32_16x16x4_f32(
                false, a, false, b, (short)0, acc[t], false, false);
        }
    }

    // D layout: lanes 0-15 -> M = tile*16 + v, N = lane; lanes 16-31 -> M = tile*16 + 8 + v
    const int mbase = tile * 16 + ((lane >> 4) * 8);
#pragma unroll
    for (int t = 0; t < 8; ++t) {
#pragma unroll
        for (int v = 0; v < 8; ++v) {
            h[(size_t)(mbase + v) * HIDDEN + t * 16 + n] = acc[t][v];
        }
    }
}

// ---------------- layer-1 scatter: wave per edge, lane covers 4 dims --------

__global__ void gcn_scatter1(const int* __restrict__ rows,
                             const int* __restrict__ cols,
                             const float* __restrict__ dinv,
                             const float* __restrict__ h,
                             float* __restrict__ agg) {
    const int e = blockIdx.x * 8 + (threadIdx.x >> 5);
    if (e >= N_EDGES) return;
    const int lane = threadIdx.x & 31;
    const int r = rows[e];
    const int c = cols[e];
    const float w = dinv[r] * dinv[c];
    const float4 hv = *(const float4*)(h + (size_t)r * HIDDEN + lane * 4);
    float* dst = agg + (size_t)c * HIDDEN + lane * 4;
    atomicAdd(dst + 0, hv.x * w);
    atomicAdd(dst + 1, hv.y * w);
    atomicAdd(dst + 2, hv.z * w);
    atomicAdd(dst + 3, hv.w * w);
}

// ---------------- bias + relu (in place on agg) ----------------

__global__ void gcn_bias_relu(float* __restrict__ agg, const float* __restrict__ b1) {
    int i = blockIdx.x * blockDim.x + threadIdx.x;
    if (i < N_NODES * HIDDEN) {
        float v = agg[i] + b1[i & (HIDDEN - 1)];
        agg[i] = fmaxf(v, 0.0f);
    }
}

// ---------------- GEMM2: z[N,10] = hrelu[N,128] @ W2[128,10] (fp32 WMMA) ----

__global__ void gcn_gemm2_wmma(const float* __restrict__ hrelu,
                               const float* __restrict__ W2,
                               float* __restrict__ z) {
    const int lane = threadIdx.x & 31;
    const int wave = threadIdx.x >> 5;
    const int tile = blockIdx.x * 8 + wave;
    const int nTiles = N_NODES / 16;
    if (tile >= nTiles) return;

    const int m    = lane & 15;
    const int koff = (lane >> 4) * 2;
    const int n    = lane & 15;
    const bool nvalid = (n < N_CLASSES);

    v8f acc = {};
    const float* hrow = hrelu + (size_t)(tile * 16 + m) * HIDDEN;

    for (int k0 = 0; k0 < HIDDEN; k0 += 4) {
        v2f a = *(const v2f*)(hrow + k0 + koff);
        v2f b;
        b.x = nvalid ? W2[(k0 + koff)     * N_CLASSES + n] : 0.0f;
        b.y = nvalid ? W2[(k0 + koff + 1) * N_CLASSES + n] : 0.0f;
        acc = __builtin_amdgcn_wmma_f32_16x16x4_f32(
            false, a, false, b, (short)0, acc, false, false);
    }

    const int mbase = tile * 16 + ((lane >> 4) * 8);
    if (nvalid) {
#pragma unroll
        for (int v = 0; v < 8; ++v) {
            z[(size_t)(mbase + v) * N_CLASSES + n] = acc[v];
        }
    }
}

// ---------------- output init with b2, then layer-2 scatter ----------------

__global__ void gcn_out_init(float* __restrict__ out, const float* __restrict__ b2) {
    int i = blockIdx.x * blockDim.x + threadIdx.x;
    if (i < N_NODES * N_CLASSES) {
        out[i] = b2[i % N_CLASSES];
    }
}

__global__ void gcn_scatter2(const int* __restrict__ rows,
                             const int* __restrict__ cols,
                             const float* __restrict__ dinv,
                             const float* __restrict__ z,
                             float* __restrict__ out) {
    int e = blockIdx.x * blockDim.x + threadIdx.x;
    if (e >= N_EDGES) return;
    const int r = rows[e];
    const int c = cols[e];
    const float w = dinv[r] * dinv[c];
    const float* zr = z + (size_t)r * N_CLASSES;
    float* dst = out + (size_t)c * N_CLASSES;
#pragma unroll
    for (int d = 0; d < N_CLASSES; ++d) {
        atomicAdd(dst + d, zr[d] * w);
    }
}

// ---------------------------------------------------------------------------

extern "C" void kernel_launch(void* const* d_in, const int* in_sizes, int n_in,
                              void* d_out, int out_size, void* d_ws, size_t ws_size,
                              hipStream_t stream) {
    const float* x   = (const float*)d_in[0];
    const int*   ei  = (const int*)d_in[1];     // edge_index, shape (2, E)
    const float* W1  = (const float*)d_in[2];
    const float* b1  = (const float*)d_in[3];
    const float* W2  = (const float*)d_in[4];
    const float* b2  = (const float*)d_in[5];
    float* out = (float*)d_out;

    const int* rows = ei;             // edge_index[0]
    const int* cols = ei + N_EDGES;   // edge_index[1]

    // Workspace layout (floats): dinv[N] | h[N*128] | agg[N*128] | z[N*10]
    float* dinv = (float*)d_ws;
    float* h    = dinv + N_NODES;                 // offset 400000 B (16B aligned)
    float* agg  = h + (size_t)N_NODES * HIDDEN;
    float* z    = agg + (size_t)N_NODES * HIDDEN;

    // Zero the accumulation buffers (graph-capturable memset nodes).
    hipMemsetAsync(dinv, 0, sizeof(float) * N_NODES, stream);
    hipMemsetAsync(agg, 0, sizeof(float) * (size_t)N_NODES * HIDDEN, stream);

    // degree + d^{-1/2}
    gcn_deg_kernel<<<(N_EDGES + 255) / 256, 256, 0, stream>>>(cols, dinv);
    gcn_dinv_kernel<<<(N_NODES + 255) / 256, 256, 0, stream>>>(dinv);

    // layer 1
    const int nTiles = N_NODES / 16;              // 6250
    const int gemmBlocks = (nTiles + 7) / 8;      // 8 waves (256 thr) per block
    gcn_gemm1_wmma<<<gemmBlocks, 256, 0, stream>>>(x, W1, h);
    gcn_scatter1<<<N_EDGES / 8, 256, 0, stream>>>(rows, cols, dinv, h, agg);
    gcn_bias_relu<<<(N_NODES * HIDDEN + 255) / 256, 256, 0, stream>>>(agg, b1);

    // layer 2
    gcn_gemm2_wmma<<<gemmBlocks, 256, 0, stream>>>(agg, W2, z);
    gcn_out_init<<<(N_NODES * N_CLASSES + 255) / 256, 256, 0, stream>>>(out, b2);
    gcn_scatter2<<<(N_EDGES + 255) / 256, 256, 0, stream>>>(rows, cols, dinv, z, out);
}